// local_graph_creator_5574867550488
// MI455X (gfx1250) — compile-verified
//
#include <hip/hip_runtime.h>
#include <math.h>

// ---------------------------------------------------------------------------
// Problem constants (from reference): N=10000 rows, D=64 features, top-K=20
// ---------------------------------------------------------------------------
#define NROWS 10000
#define DFEAT 64
#define ALPHA 3.0f
#define TOPK  20
#define NTILE 625          // NROWS / 16
#define WAVES_PER_BLOCK 8  // 256 threads, wave32

typedef __attribute__((ext_vector_type(2))) float v2f;
typedef __attribute__((ext_vector_type(8))) float v8f;

// ---------------------------------------------------------------------------
// Kernel 1: vec1[i,d] = tanh(ALPHA * (sum_k emb[idx[i],k] * fc1_w[d,k] + b[d]))
// One 64-thread block per row; row broadcast via LDS. Tiny (~82 MFLOP).
// ---------------------------------------------------------------------------
__global__ void lgc_vec1_kernel(const int* __restrict__ idx,
                                const float* __restrict__ emb,
                                const float* __restrict__ w,
                                const float* __restrict__ b,
                                float* __restrict__ vec1) {
    __shared__ float srow[DFEAT];
    const int i = blockIdx.x;
    const int d = threadIdx.x;            // 0..63
    const int r = idx[i];
    srow[d] = emb[(size_t)r * DFEAT + d];
    __syncthreads();
    const float* wrow = w + (size_t)d * DFEAT;   // fc1_w[d, :]
    float s = 0.0f;
#pragma unroll 8
    for (int k = 0; k < DFEAT; ++k) s += srow[k] * wrow[k];
    vec1[(size_t)i * DFEAT + d] = tanhf(ALPHA * (s + b[d]));
}

// ---------------------------------------------------------------------------
// Kernel 2: adj = relu(tanh(ALPHA * (vec1 @ gEmb^T - gEmb @ vec1^T)))
// One wave per 16x16 output tile; two f32 WMMA accumulation chains over K=64
// (16 k-steps of V_WMMA_F32_16X16X4_F32 each). f32 kept on purpose: the op is
// HBM-bound (400MB store), not math-bound, and top-k selection is precision-
// sensitive.
//
// Fragment layouts per CDNA5 ISA 7.12.2 (32-bit A 16x4 / B 4x16):
//   A: lanes 0-15 -> M=lane, {K=kk+0, kk+1}; lanes 16-31 -> M=lane-16, {kk+2, kk+3}
//   B: lanes 0-15 -> N=lane, {K=kk+0, kk+1}; lanes 16-31 -> N=lane-16, {kk+2, kk+3}
//   C/D: vgpr v -> M = v + 8*(lane>=16), N = lane%16
// ---------------------------------------------------------------------------
__global__ void lgc_score_kernel(const float* __restrict__ vec1,
                                 const float* __restrict__ gEmb,
                                 float* __restrict__ out) {
    const int ib   = blockIdx.x;                         // row tile 0..624
    const int wave = threadIdx.x >> 5;
    const int jb   = blockIdx.y * WAVES_PER_BLOCK + wave; // col tile
    if (jb >= NTILE) return;                              // wave-uniform exit

    const int lane = threadIdx.x & 31;
    const int half = lane >> 4;          // 0: lanes 0-15, 1: lanes 16-31
    const int l16  = lane & 15;
    const int ko   = half * 2;           // K sub-offset for this lane half

    const int arow = ib * 16 + l16;      // source row for A fragments
    const int bcol = jb * 16 + l16;      // source row (= output column) for B
    const float* v1a = vec1 + (size_t)arow * DFEAT;
    const float* gea = gEmb + (size_t)arow * DFEAT;
    const float* v1b = vec1 + (size_t)bcol * DFEAT;
    const float* geb = gEmb + (size_t)bcol * DFEAT;

    v8f c1 = {};   // vec1 @ gEmb^T tile
    v8f c2 = {};   // gEmb @ vec1^T tile
#pragma unroll
    for (int kk = 0; kk < DFEAT; kk += 4) {
        v2f a1 = { v1a[kk + ko], v1a[kk + ko + 1] };
        v2f b1 = { geb[kk + ko], geb[kk + ko + 1] };
        c1 = __builtin_amdgcn_wmma_f32_16x16x4_f32(
                 false, a1, false, b1, (short)0, c1, false, false);
        v2f a2 = { gea[kk + ko], gea[kk + ko + 1] };
        v2f b2 = { v1b[kk + ko], v1b[kk + ko + 1] };
        c2 = __builtin_amdgcn_wmma_f32_16x16x4_f32(
                 false, a2, false, b2, (short)0, c2, false, false);
    }

    // Epilogue: a = c1 - c2 ; adj = relu(tanh(ALPHA*a)) ; coalesced stores
    const size_t base = (size_t)(ib * 16 + half * 8) * NROWS + (size_t)jb * 16 + l16;
#pragma unroll
    for (int v = 0; v < 8; ++v) {
        float aval = c1[v] - c2[v];
        float adj  = tanhf(ALPHA * aval);
        adj = adj > 0.0f ? adj : 0.0f;
        out[base + (size_t)v * NROWS] = adj;
    }
}

// ---------------------------------------------------------------------------
// Kernel 3: in-place per-row top-K filter. One 256-thread block per row.
// Row (40KB) staged in LDS (320KB/WGP on CDNA5 -> comfortable). 20 rounds of
// tie-aware argmax (lower index wins ties, matching jax.lax.top_k), then the
// row is rewritten keeping only the K winners.
// ---------------------------------------------------------------------------
#define TPB 256
__global__ void lgc_topk_kernel(float* __restrict__ out) {
    __shared__ float srow[NROWS];
    __shared__ float redv[TPB];
    __shared__ int   redi[TPB];
    __shared__ float topv[TOPK];
    __shared__ int   topi[TOPK];

    const int row = blockIdx.x;
    float* rp = out + (size_t)row * NROWS;

    for (int c = threadIdx.x; c < NROWS; c += TPB) srow[c] = rp[c];
    __syncthreads();

    for (int it = 0; it < TOPK; ++it) {
        // per-thread argmax over strided chunk (ascending c => strict '>' keeps
        // the lowest index among equal values)
        float mv = -2.0f; int mi = NROWS;
        for (int c = threadIdx.x; c < NROWS; c += TPB) {
            float v = srow[c];
            if (v > mv) { mv = v; mi = c; }
        }
        redv[threadIdx.x] = mv; redi[threadIdx.x] = mi;
        __syncthreads();
        // tree reduce, lower index wins ties
        for (int s = TPB / 2; s > 0; s >>= 1) {
            if (threadIdx.x < s) {
                float ov = redv[threadIdx.x + s]; int oi = redi[threadIdx.x + s];
                float cv = redv[threadIdx.x];     int ci = redi[threadIdx.x];
                if (ov > cv || (ov == cv && oi < ci)) {
                    redv[threadIdx.x] = ov; redi[threadIdx.x] = oi;
                }
            }
            __syncthreads();
        }
        if (threadIdx.x == 0) {
            topv[it] = redv[0];
            topi[it] = redi[0];
            srow[redi[0]] = -1.0f;   // below any relu'd value; excluded next round
        }
        __syncthreads();
    }

    // rewrite row: keep only the K winners (original values), zero elsewhere
    for (int c = threadIdx.x; c < NROWS; c += TPB) {
        float v = 0.0f;
#pragma unroll
        for (int t = 0; t < TOPK; ++t) {
            if (topi[t] == c) v = topv[t];
        }
        rp[c] = v;
    }
}

// ---------------------------------------------------------------------------
// Launch: inputs in setup_inputs() order: idx, gEmb, emb_weight, fc1_w, fc1_b
// d_out (N*N f32) doubles as the adj scratch; vec1 lives in d_ws (2.56 MB).
// ---------------------------------------------------------------------------
extern "C" void kernel_launch(void* const* d_in, const int* in_sizes, int n_in,
                              void* d_out, int out_size, void* d_ws, size_t ws_size,
                              hipStream_t stream) {
    const int*   idx        = (const int*)  d_in[0];
    const float* gEmb       = (const float*)d_in[1];
    const float* emb_weight = (const float*)d_in[2];
    const float* fc1_w      = (const float*)d_in[3];
    const float* fc1_b      = (const float*)d_in[4];
    float* out  = (float*)d_out;
    float* vec1 = (float*)d_ws;   // NROWS * DFEAT floats = 2.56 MB

    lgc_vec1_kernel<<<NROWS, DFEAT, 0, stream>>>(idx, emb_weight, fc1_w, fc1_b, vec1);

    dim3 g2(NTILE, (NTILE + WAVES_PER_BLOCK - 1) / WAVES_PER_BLOCK);
    lgc_score_kernel<<<g2, 32 * WAVES_PER_BLOCK, 0, stream>>>(vec1, gEmb, out);

    lgc_topk_kernel<<<NROWS, TPB, 0, stream>>>(out);
}